// GraphNN_57947698757816
// MI455X (gfx1250) — compile-verified
//
#include <hip/hip_runtime.h>

// CDNA5 / gfx1250. fp32 WMMA (v_wmma_f32_16x16x4_f32) + TDM tensor_load_to_lds
// for activation tiles (TENSORcnt) + global_load_async_to_lds_b128 for weight
// tiles (ASYNCcnt), both double-buffered against the WMMA stages.
typedef __attribute__((ext_vector_type(2))) float v2f;
typedef __attribute__((ext_vector_type(4))) float f4;
typedef __attribute__((ext_vector_type(8))) float v8f;
typedef unsigned int v4u __attribute__((ext_vector_type(4)));
typedef int v8i __attribute__((ext_vector_type(8)));
typedef int v4i __attribute__((ext_vector_type(4)));

#define BSZ   262144
#define BROWS 128      // batch rows per block
#define LDA   68       // padded LDS row stride (floats) -> conflict-free frag reads
#define LDW   68       // same padding for weight tile (row-major, no transpose)
#define ASIZE (BROWS * LDA)   // 8704 floats per A buffer (incl. trailing TDM pad)
#define WSIZE (64 * LDW)      // 4352 floats per W buffer

__device__ __forceinline__ unsigned lds_off(const void* p) {
    // LDS aperture: low 32 bits of a generic pointer are the LDS byte address
    return (unsigned)(uintptr_t)p;
}

// ---------------------------------------------------------------------------
// TDM: DMA a 128x64 f32 tile (row stride 64) from global into LDS, with the
// TDM padding feature inserting 4 DWORDs after every 64 DWORDs -> LDS row
// stride 68 = exactly our bank-conflict-free layout.  2-D descriptor, groups
// 2..4 zero (tile_dim2 = 0 -> unused).  Tracked on TENSORcnt.
// This toolchain's builtin takes 6 args (g0, g1, g2, g3, g4, cpol).
// ---------------------------------------------------------------------------
__device__ __forceinline__ void tdm_load_A(const float* gsrc, unsigned lds_addr) {
    unsigned long long ga = (unsigned long long)(uintptr_t)gsrc;
    v4u g0;
    g0.x = 1u;                                                // count=1, user mode
    g0.y = lds_addr;                                          // lds_addr [63:32]
    g0.z = (unsigned)(ga & 0xffffffffu);                      // global_addr lo
    g0.w = (unsigned)((ga >> 32) & 0x01ffffffu) | (2u << 30); // ga hi | type=2
    v8i g1;
    g1[0] = (2 << 16)        // data_size = 4B
          | (1 << 20)        // pad_enable
          | (5 << 22)        // pad_interval: 64 DWORDs
          | (3 << 25);       // pad_amount: 4 DWORDs
    g1[1] = (64 << 16);      // tensor_dim0 = 64 (bits 79:48)
    g1[2] = (BROWS << 16);   // tensor_dim1 = 128 (bits 111:80)
    g1[3] = (64 << 16);      // tile_dim0 = 64 (bits 127:112)
    g1[4] = BROWS;           // tile_dim1 = 128 (bits 143:128), tile_dim2 = 0
    g1[5] = 64;              // tensor_dim0_stride = 64 (bits 207:160 lo)
    g1[6] = 0;
    g1[7] = 0;
    v4i gz4 = {0, 0, 0, 0};
    v8i gz8 = {0, 0, 0, 0, 0, 0, 0, 0};
    __builtin_amdgcn_tensor_load_to_lds(g0, g1, gz4, gz4, gz8, 0);
}

// ---------------------------------------------------------------------------
// Async-copy one 64x64 f32 weight tile into LDS (row-major, stride LDW).
// 1024 16B chunks split across 256 threads.  Tracked on ASYNCcnt.
// ---------------------------------------------------------------------------
__device__ __forceinline__ void async_copy_W_tile(const float* Wp,
                                                  unsigned lds_base, int tid) {
    unsigned long long sbase = (unsigned long long)(uintptr_t)Wp;
    #pragma unroll
    for (int it = 0; it < 4; ++it) {
        int idx = it * 256 + tid;                 // chunk id 0..1023
        int v = idx >> 4;                         // row
        int c = (idx & 15) << 2;                  // col (floats)
        unsigned lds  = lds_base + (unsigned)((v * LDW + c) * 4);
        unsigned goff = (unsigned)idx * 16u;
        asm volatile("global_load_async_to_lds_b128 %0, %1, %2"
                     :: "v"(lds), "v"(goff), "s"(sbase) : "memory");
    }
}

// ---------------------------------------------------------------------------
// GEMM stage: Hout[k][b][v] = sum_p sum_w A[p][b][w] * W[p][k][v][w] + bias[k][v]
// plus per-block partial sum-of-squares for the Frobenius normalization.
// Weights are PRE-SCALED by 1/||h_p|| where applicable.
// Fully unrolled over (p, k) so LDS double-buffer bases are compile-time
// constants -> ds_load immediate offsets instead of per-address VALU adds.
// ---------------------------------------------------------------------------
template<int P, int KN>
__global__ __launch_bounds__(256)
void gemm_norm_kernel(const float* __restrict__ A,     // [P][B][64]
                      const float* __restrict__ W,     // [P][KN][64][64] (v-major)
                      const float* __restrict__ bias,  // [KN][64]
                      float* __restrict__ Hout,        // [KN][B][64]
                      float* __restrict__ partial)     // [KN][gridDim.x]
{
    __shared__ float lsA[2][ASIZE];
    __shared__ float lsW[2][WSIZE];
    __shared__ float red[KN][8];

    const int tid   = threadIdx.x;
    const int lane  = tid & 31;
    const int wave  = tid >> 5;       // 0..7
    const int laneM = lane & 15;
    const int laneH = lane >> 4;      // 0/1
    const long m0   = (long)blockIdx.x * BROWS;

    const unsigned lsA_base = lds_off(&lsA[0][0]);
    const unsigned lsW_base = lds_off(&lsW[0][0]);

    v8f d[KN][4] = {};                // [node][n-tile] 16x16 f32 accumulators

    // ---- prologue: prime buffer 0 on both DMA paths ----
    if (wave == 0) tdm_load_A(A + m0 * 64, lsA_base);
    async_copy_W_tile(W, lsW_base, tid);
    asm volatile("s_wait_asynccnt 0x0" ::: "memory");
    __builtin_amdgcn_s_wait_tensorcnt((short)0);   // no-op for waves != 0
    __syncthreads();

    #pragma unroll
    for (int p = 0; p < P; ++p) {
        #pragma unroll
        for (int k = 0; k < KN; ++k) {
            const int s = p * KN + k;
            // prefetch next weight tile (ASYNCcnt) into the other buffer
            if (s + 1 < P * KN) {
                const int np = (k + 1 < KN) ? p : p + 1;
                const int nk = (k + 1 < KN) ? k + 1 : 0;
                async_copy_W_tile(W + ((long)np * KN + nk) * 4096,
                                  lsW_base + (unsigned)(((s + 1) & 1) * (WSIZE * 4)),
                                  tid);
            }
            // prefetch next activation tile via TDM (TENSORcnt); stays in
            // flight across all KN stages of this p since counters are split
            if (k == 0 && p + 1 < P && wave == 0)
                tdm_load_A(A + (long)(p + 1) * BSZ * 64 + m0 * 64,
                           lsA_base + (unsigned)(((p + 1) & 1) * (ASIZE * 4)));

            // ---- 16 K-steps of v_wmma_f32_16x16x4_f32 over 4 N-tiles ----
            const int ab = p & 1;     // compile-time after unroll
            const int wb = s & 1;     // compile-time after unroll
            const int r0 = wave * 16;
            #pragma unroll
            for (int kk = 0; kk < 16; ++kk) {
                // A frag 16x4: lanes0-15 K={4kk,4kk+1}, lanes16-31 K={4kk+2,4kk+3}
                v2f a = *(const v2f*)&lsA[ab][(r0 + laneM) * LDA + kk * 4 + laneH * 2];
                #pragma unroll
                for (int nt = 0; nt < 4; ++nt) {
                    // B frag: b = { W[v][w0], W[v][w0+1] }, contiguous in row-major
                    v2f b = *(const v2f*)&lsW[wb][(nt * 16 + laneM) * LDW
                                                  + kk * 4 + laneH * 2];
                    d[k][nt] = __builtin_amdgcn_wmma_f32_16x16x4_f32(
                        false, a, false, b, (short)0, d[k][nt], false, false);
                }
            }
            // next-stage W must be resident; A only at p boundaries
            asm volatile("s_wait_asynccnt 0x0" ::: "memory");
            if (k == KN - 1 && p + 1 < P)
                __builtin_amdgcn_s_wait_tensorcnt((short)0);
            __syncthreads();
        }
    }

    // ---- epilogue: bias, sum-of-squares, store ----
    #pragma unroll
    for (int k = 0; k < KN; ++k) {
        float ss = 0.f;
        const float* bk = bias + k * 64;
        float* ok = Hout + (long)k * (long)BSZ * 64
                         + (m0 + wave * 16 + laneH * 8) * 64;
        #pragma unroll
        for (int nt = 0; nt < 4; ++nt) {
            float bv = bk[nt * 16 + laneM];
            #pragma unroll
            for (int j = 0; j < 8; ++j) {     // D: vgpr j -> row j (+8 for hi half)
                float v = d[k][nt][j] + bv;
                ss += v * v;
                ok[j * 64 + nt * 16 + laneM] = v;
            }
        }
        #pragma unroll
        for (int off = 16; off > 0; off >>= 1) ss += __shfl_xor(ss, off, 32);
        if (lane == 0) red[k][wave] = ss;
    }
    __syncthreads();
    if (tid < KN) {
        float t = 0.f;
        #pragma unroll
        for (int i = 0; i < 8; ++i) t += red[tid][i];
        partial[tid * gridDim.x + blockIdx.x] = t;
    }
}

// Deterministic tree reduction of per-block partials -> sumsq[k]
__global__ void reduce_ss_kernel(const float* __restrict__ partial,
                                 float* __restrict__ ss, int nblocks) {
    __shared__ float acc[256];
    float t = 0.f;
    for (int i = threadIdx.x; i < nblocks; i += 256)
        t += partial[blockIdx.x * nblocks + i];
    acc[threadIdx.x] = t;
    __syncthreads();
    for (int s = 128; s > 0; s >>= 1) {
        if (threadIdx.x < s) acc[threadIdx.x] += acc[threadIdx.x + s];
        __syncthreads();
    }
    if (threadIdx.x == 0) ss[blockIdx.x] = acc[0];
}

// Wdst[i] = Wsrc[i] * 1/||h_p||  (fold predecessor normalization into weights)
__global__ void scale_w_kernel(const float* __restrict__ Wsrc,
                               const float* __restrict__ ss,
                               float* __restrict__ Wdst, int Kn, int n) {
    int i = blockIdx.x * blockDim.x + threadIdx.x;
    if (i < n) {
        int p = i / (Kn * 4096);
        Wdst[i] = Wsrc[i] * rsqrtf(ss[p]);
    }
}

// bdst[k][v] = sum_p bsrc[p][k][v]
__global__ void combine_bias_kernel(const float* __restrict__ bsrc,
                                    float* __restrict__ bdst, int P, int Kn) {
    int i = blockIdx.x * blockDim.x + threadIdx.x;
    if (i < Kn * 64) {
        int k = i >> 6, v = i & 63;
        float s = 0.f;
        for (int p = 0; p < P; ++p) s += bsrc[(p * Kn + k) * 64 + v];
        bdst[i] = s;
    }
}

// out *= 1/||out||  (final global normalization)
__global__ void finalize_kernel(float* __restrict__ out,
                                const float* __restrict__ ss, int n4) {
    int i = blockIdx.x * blockDim.x + threadIdx.x;
    float s = rsqrtf(ss[0]);
    if (i < n4) {
        f4 v = ((const f4*)out)[i];
        v.x *= s; v.y *= s; v.z *= s; v.w *= s;
        ((f4*)out)[i] = v;
    }
}

extern "C" void kernel_launch(void* const* d_in, const int* in_sizes, int n_in,
                              void* d_out, int out_size, void* d_ws, size_t ws_size,
                              hipStream_t stream) {
    const float* x     = (const float*)d_in[0];   // [B][64]
    const float* W_in  = (const float*)d_in[1];   // [4][64][64]
    const float* b_in  = (const float*)d_in[2];   // [4][64]
    const float* W_mid = (const float*)d_in[3];   // [3][4][4][64][64]
    const float* b_mid = (const float*)d_in[4];   // [3][4][4][64]
    const float* W_out = (const float*)d_in[5];   // [4][64][64]
    const float* b_out = (const float*)d_in[6];   // [4][64]
    float* out = (float*)d_out;                   // [B][64]

    const int Kn = 4;
    const int nblk = BSZ / BROWS;                 // 2048

    // workspace layout
    char* ws = (char*)d_ws;
    const size_t szH = (size_t)Kn * BSZ * 64 * sizeof(float);   // 256 MiB
    float* Ha    = (float*)(ws);
    float* Hb    = (float*)(ws + szH);
    float* Wsc   = (float*)(ws + 2 * szH);                       // up to 4*4*4096 f
    float* bcomb = (float*)(ws + 2 * szH + 262144);              // up to 4*64 f
    float* ssb   = (float*)(ws + 2 * szH + 262144 + 1024);       // 16 f: slot0/1/out
    float* part  = (float*)(ws + 2 * szH + 262144 + 1024 + 64);  // [4][2048]

    dim3 blk(256);

    // stage 0: input -> layer 1 (no input normalization)
    gemm_norm_kernel<1, 4><<<dim3(nblk), blk, 0, stream>>>(x, W_in, b_in, Ha, part);
    reduce_ss_kernel<<<4, 256, 0, stream>>>(part, ssb + 0, nblk);

    // mid transitions
    float* cur = Ha; float* nxt = Hb;
    for (int l = 0; l < 3; ++l) {
        const float* Wl = W_mid + (size_t)l * Kn * Kn * 4096;
        const float* bl = b_mid + (size_t)l * Kn * Kn * 64;
        float* ssPrev = ssb + (l & 1) * 4;
        float* ssCur  = ssb + ((l + 1) & 1) * 4;
        scale_w_kernel<<<(Kn * Kn * 4096 + 255) / 256, 256, 0, stream>>>(
            Wl, ssPrev, Wsc, Kn, Kn * Kn * 4096);
        combine_bias_kernel<<<1, 256, 0, stream>>>(bl, bcomb, Kn, Kn);
        gemm_norm_kernel<4, 4><<<dim3(nblk), blk, 0, stream>>>(cur, Wsc, bcomb, nxt, part);
        reduce_ss_kernel<<<4, 256, 0, stream>>>(part, ssCur, nblk);
        float* t = cur; cur = nxt; nxt = t;
    }

    // output stage: prev scales live in slot (3&1)==1
    float* ssPrev = ssb + 4;
    float* ssOut  = ssb + 8;
    scale_w_kernel<<<(Kn * 4096 + 255) / 256, 256, 0, stream>>>(
        W_out, ssPrev, Wsc, 1, Kn * 4096);
    combine_bias_kernel<<<1, 64, 0, stream>>>(b_out, bcomb, Kn, 1);
    gemm_norm_kernel<4, 1><<<dim3(nblk), blk, 0, stream>>>(cur, Wsc, bcomb, out, part);
    reduce_ss_kernel<<<1, 256, 0, stream>>>(part, ssOut, nblk);

    const int n4 = BSZ * 64 / 4;
    finalize_kernel<<<(n4 + 255) / 256, 256, 0, stream>>>(out, ssOut, n4);
}